// GATconv_hyperkcp_47717086659279
// MI455X (gfx1250) — compile-verified
//
#include <hip/hip_runtime.h>
#include <hip/hip_bf16.h>

#define NV   100000
#define NE   20000
#define NNZC 800000
#define DD   128

typedef __attribute__((ext_vector_type(2))) float v2f;
typedef __attribute__((ext_vector_type(8))) float v8f;

// ---------------------------------------------------------------------------
// 1) H = X @ W + b  via fp32 WMMA 16x16x4. One wave computes one 16x16 tile.
//    A (16x4):  lanes 0-15 hold {K=k,k+1} for M=lane;  lanes 16-31 hold {K=k+2,k+3}.
//    B (4x16):  VGPR r, lanes 0-15 -> K=r, lanes 16-31 -> K=r+2, N=lane%16.
//    C (16x16): VGPR r, lanes 0-15 -> M=r, lanes 16-31 -> M=r+8, N=lane%16.
// ---------------------------------------------------------------------------
__global__ __launch_bounds__(128) void k_gemm_wmma(const float* __restrict__ X,
                                                   const float* __restrict__ W,
                                                   const float* __restrict__ bvec,
                                                   float* __restrict__ H) {
    const int wave  = blockIdx.x * (blockDim.x >> 5) + (threadIdx.x >> 5);
    const int tileM = wave >> 3;        // 6250 tiles in M
    const int tileN = wave & 7;         // 8 tiles in N
    const int lane  = threadIdx.x & 31;
    const int half  = lane >> 4;        // 0 or 1
    const int l16   = lane & 15;

    const float* arow = X + (size_t)(tileM * 16 + l16) * DD;
    const int    colB = tileN * 16 + l16;

    v8f c = {};
#pragma unroll 4
    for (int k = 0; k < DD; k += 4) {
        v2f a, b;
        a[0] = arow[k + 2 * half];
        a[1] = arow[k + 2 * half + 1];
        b[0] = W[(size_t)(k + 2 * half)     * DD + colB];
        b[1] = W[(size_t)(k + 2 * half + 1) * DD + colB];
        c = __builtin_amdgcn_wmma_f32_16x16x4_f32(false, a, false, b,
                                                  (short)0, c, false, false);
    }

    const float bias    = bvec[colB];
    const int   rowBase = tileM * 16 + half * 8;
#pragma unroll
    for (int r = 0; r < 8; ++r)
        H[(size_t)(rowBase + r) * DD + colB] = c[r] + bias;
}

// ---------------------------------------------------------------------------
// 2) Per-column sum / sumsq partials for batchnorm (coalesced row sweeps).
// ---------------------------------------------------------------------------
__global__ __launch_bounds__(128) void k_stats(const float* __restrict__ H,
                                               float* __restrict__ colsum,
                                               float* __restrict__ colsq,
                                               int rowsPerBlock) {
    const int t  = threadIdx.x;                      // column 0..127
    int r0 = blockIdx.x * rowsPerBlock;
    int r1 = r0 + rowsPerBlock; if (r1 > NV) r1 = NV;
    float s = 0.f, q = 0.f;
    for (int r = r0; r < r1; ++r) {
        float v = H[(size_t)r * DD + t];
        s += v; q += v * v;
    }
    atomicAdd(&colsum[t], s);
    atomicAdd(&colsq[t],  q);
}

// 3) Fold stats + gamma/beta into per-column scale/shift.
__global__ void k_bn_finalize(const float* __restrict__ colsum,
                              const float* __restrict__ colsq,
                              const float* __restrict__ gamma,
                              const float* __restrict__ beta,
                              float* __restrict__ scale,
                              float* __restrict__ shift) {
    const int t = threadIdx.x;
    const float invN = 1.0f / (float)NV;
    float mean = colsum[t] * invN;
    float var  = colsq[t] * invN - mean * mean;      // biased var (matches jnp.var)
    float sc   = gamma[t] * rsqrtf(var + 1e-5f);
    scale[t] = sc;
    shift[t] = beta[t] - mean * sc;
}

// 4) Apply BN in place, float4-vectorized.
__global__ __launch_bounds__(256) void k_bn_apply(float* __restrict__ H,
                                                  const float* __restrict__ scale,
                                                  const float* __restrict__ shift) {
    const int i = blockIdx.x * blockDim.x + threadIdx.x;   // float4 index
    if (i >= NV * DD / 4) return;
    const int d4 = i & (DD / 4 - 1);
    float4 h  = ((float4*)H)[i];
    float4 sc = ((const float4*)scale)[d4];
    float4 sh = ((const float4*)shift)[d4];
    h.x = fmaf(h.x, sc.x, sh.x);
    h.y = fmaf(h.y, sc.y, sh.y);
    h.z = fmaf(h.z, sc.z, sh.z);
    h.w = fmaf(h.w, sc.w, sh.w);
    ((float4*)H)[i] = h;
}

// ---------------------------------------------------------------------------
// 5) Hyperedge membership counts.
// ---------------------------------------------------------------------------
__global__ __launch_bounds__(256) void k_count(const int* __restrict__ e_idx,
                                               float* __restrict__ cnt) {
    const int i = blockIdx.x * blockDim.x + threadIdx.x;
    if (i >= NNZC) return;
    atomicAdd(&cnt[e_idx[i]], 1.0f);
}

// 6) v2e scatter-add: wave per incidence entry, float4 per lane, L2 atomics.
__global__ __launch_bounds__(256) void k_v2e(const float* __restrict__ H,
                                             const int* __restrict__ v_idx,
                                             const int* __restrict__ e_idx,
                                             float* __restrict__ Y) {
    const int entry = blockIdx.x * (blockDim.x >> 5) + (threadIdx.x >> 5);
    if (entry >= NNZC) return;
    const int lane = threadIdx.x & 31;
    const int v = v_idx[entry];
    const int e = e_idx[entry];
    float4 h = ((const float4*)(H + (size_t)v * DD))[lane];
    float* yp = Y + (size_t)e * DD + lane * 4;
    atomicAdd(yp + 0, h.x);
    atomicAdd(yp + 1, h.y);
    atomicAdd(yp + 2, h.z);
    atomicAdd(yp + 3, h.w);
}

// 7) Per-edge: Y /= max(cnt,1); alpha = tanh(Y . W_gate);
//    s = clip(leakyrelu_0.2(alpha), 0, 5) precomputed per edge.
__global__ __launch_bounds__(256) void k_edge(float* __restrict__ Y,
                                              const float* __restrict__ cnt,
                                              const float* __restrict__ Wg,
                                              float* __restrict__ sclip) {
    const int e = blockIdx.x * (blockDim.x >> 5) + (threadIdx.x >> 5);
    if (e >= NE) return;
    const int lane = threadIdx.x & 31;
    const float inv = 1.0f / fmaxf(cnt[e], 1.0f);
    float4* yp = (float4*)(Y + (size_t)e * DD);
    float4 y = yp[lane];
    y.x *= inv; y.y *= inv; y.z *= inv; y.w *= inv;
    yp[lane] = y;
    float4 w = ((const float4*)Wg)[lane];
    float p = y.x * w.x + y.y * w.y + y.z * w.z + y.w * w.w;
#pragma unroll
    for (int m = 16; m >= 1; m >>= 1) p += __shfl_xor(p, m, 32);
    if (lane == 0) {
        float al = tanhf(p);
        float s  = al >= 0.f ? al : 0.2f * al;     // LeakyReLU(0.2)
        s = fminf(fmaxf(s, 0.0f), 5.0f);           // clamp [0,5]
        sclip[e] = s;
    }
}

// 8) e2v scatter-add with group weight a_g folded in; single fused accumulator.
__global__ __launch_bounds__(256) void k_e2v(const float* __restrict__ Y,
                                             const float* __restrict__ sclip,
                                             const int* __restrict__ v_idx,
                                             const int* __restrict__ e_idx,
                                             const float* __restrict__ a_g,
                                             float* __restrict__ ACC) {
    const int entry = blockIdx.x * (blockDim.x >> 5) + (threadIdx.x >> 5);
    if (entry >= NNZC) return;
    const int lane = threadIdx.x & 31;
    const int v = v_idx[entry];
    const int e = e_idx[entry];
    const float w = a_g[0] * sclip[e];
    float4 y = ((const float4*)(Y + (size_t)e * DD))[lane];
    float* ap = ACC + (size_t)v * DD + lane * 4;
    atomicAdd(ap + 0, y.x * w);
    atomicAdd(ap + 1, y.y * w);
    atomicAdd(ap + 2, y.z * w);
    atomicAdd(ap + 3, y.w * w);
}

// 9) out = ACC + 0.999*H, LeakyReLU(0.01).
__global__ __launch_bounds__(256) void k_final(const float* __restrict__ ACC,
                                               const float* __restrict__ H,
                                               float* __restrict__ out) {
    const int i = blockIdx.x * blockDim.x + threadIdx.x;
    if (i >= NV * DD / 4) return;
    float4 a = ((const float4*)ACC)[i];
    float4 h = ((const float4*)H)[i];
    float4 o;
    o.x = fmaf(0.999f, h.x, a.x);
    o.y = fmaf(0.999f, h.y, a.y);
    o.z = fmaf(0.999f, h.z, a.z);
    o.w = fmaf(0.999f, h.w, a.w);
    o.x = o.x >= 0.f ? o.x : 0.01f * o.x;
    o.y = o.y >= 0.f ? o.y : 0.01f * o.y;
    o.z = o.z >= 0.f ? o.z : 0.01f * o.z;
    o.w = o.w >= 0.f ? o.w : 0.01f * o.w;
    ((float4*)out)[i] = o;
}

// ---------------------------------------------------------------------------
extern "C" void kernel_launch(void* const* d_in, const int* in_sizes, int n_in,
                              void* d_out, int out_size, void* d_ws, size_t ws_size,
                              hipStream_t stream) {
    const float* X       = (const float*)d_in[0];
    const float* W_theta = (const float*)d_in[1];
    const float* b_theta = (const float*)d_in[2];
    const float* gamma   = (const float*)d_in[3];
    const float* beta    = (const float*)d_in[4];
    const float* W_gate  = (const float*)d_in[5];
    const float* a_g[3]  = { (const float*)d_in[6], (const float*)d_in[7], (const float*)d_in[8] };
    const int*   v_idx[3]= { (const int*)d_in[9],  (const int*)d_in[11], (const int*)d_in[13] };
    const int*   e_idx[3]= { (const int*)d_in[10], (const int*)d_in[12], (const int*)d_in[14] };
    float* out = (float*)d_out;

    // workspace carve-up (256B aligned)
    size_t off = 0;
    auto carve = [&](size_t elems) {
        float* p = (float*)((char*)d_ws + off);
        off += ((elems * sizeof(float) + 255) / 256) * 256;
        return p;
    };
    float* H      = carve((size_t)NV * DD);   // 51.2 MB
    float* ACC    = carve((size_t)NV * DD);   // 51.2 MB
    float* Y      = carve((size_t)NE * DD);   // 10.2 MB (reused per group)
    float* cnt    = carve(NE);
    float* sclip  = carve(NE);
    float* stats  = carve(512);               // colsum|colsq|scale|shift
    float* colsum = stats;
    float* colsq  = stats + 128;
    float* scale  = stats + 256;
    float* shift  = stats + 384;

    hipMemsetAsync(ACC,   0, (size_t)NV * DD * sizeof(float), stream);
    hipMemsetAsync(stats, 0, 256 * sizeof(float), stream);

    // GEMM: 6250*8 = 50000 tiles, 4 waves/block
    k_gemm_wmma<<<50000 / 4, 128, 0, stream>>>(X, W_theta, b_theta, H);

    // BN stats + apply
    const int statBlocks = 256;
    const int rowsPerBlk = (NV + statBlocks - 1) / statBlocks;
    k_stats<<<statBlocks, 128, 0, stream>>>(H, colsum, colsq, rowsPerBlk);
    k_bn_finalize<<<1, 128, 0, stream>>>(colsum, colsq, gamma, beta, scale, shift);
    k_bn_apply<<<(NV * DD / 4 + 255) / 256, 256, 0, stream>>>(H, scale, shift);

    // three propagation groups into one fused accumulator
    for (int g = 0; g < 3; ++g) {
        hipMemsetAsync(Y,   0, (size_t)NE * DD * sizeof(float), stream);
        hipMemsetAsync(cnt, 0, NE * sizeof(float), stream);
        k_count<<<(NNZC + 255) / 256, 256, 0, stream>>>(e_idx[g], cnt);
        k_v2e<<<NNZC / 8, 256, 0, stream>>>(H, v_idx[g], e_idx[g], Y);
        k_edge<<<(NE + 7) / 8, 256, 0, stream>>>(Y, cnt, W_gate, sclip);
        k_e2v<<<NNZC / 8, 256, 0, stream>>>(Y, sclip, v_idx[g], e_idx[g], a_g[g], ACC);
    }

    k_final<<<(NV * DD / 4 + 255) / 256, 256, 0, stream>>>(ACC, H, out);
}